// AperiodicKNN_PyG_22179211117017
// MI455X (gfx1250) — compile-verified
//
#include <hip/hip_runtime.h>

// CDNA5 / gfx1250: wave32, WMMA f32 16x16x4.
typedef __attribute__((ext_vector_type(2))) float v2f;
typedef __attribute__((ext_vector_type(8))) float v8f;

#define KNN_K 17

__global__ __launch_bounds__(32)
void knn_wmma_kernel(const float* __restrict__ pos, int n,
                     int* __restrict__ ei_src, int* __restrict__ ei_dst,
                     float* __restrict__ ed) {
    // Per-wave transpose staging for the two 16x16 WMMA result tiles (2 KB).
    __shared__ float tile[2][16][16];

    const int lane = threadIdx.x;        // 0..31
    const int half = lane >> 4;          // 0: K=0,1 lanes ; 1: K=2,3 lanes
    const int col  = lane & 15;
    const int c0   = blockIdx.x * 32;

    // This lane's own center (for top-k bookkeeping and output).
    const int c  = c0 + lane;
    const int cc = c < n ? c : n - 1;
    const float cx = pos[3*cc+0], cy = pos[3*cc+1], cz = pos[3*cc+2];
    const float x2i = cx*cx + cy*cy + cz*cz;

    // A matrices (16x4, loaded once, reused over all candidate tiles).
    // Row m of A = (-2x_m, -2y_m, -2z_m, 1). ISA layout: lane(half=0,row=col)
    // holds (K0,K1); lane(half=1,row=col) holds (K2,K3).
    v2f a0, a1;
    {
        int ra = c0 + col;      ra = ra < n ? ra : n - 1;
        int rb = c0 + 16 + col; rb = rb < n ? rb : n - 1;
        const float ax = pos[3*ra+0], ay = pos[3*ra+1], az = pos[3*ra+2];
        const float bx = pos[3*rb+0], by = pos[3*rb+1], bz = pos[3*rb+2];
        if (half == 0) { a0.x = -2.0f*ax; a0.y = -2.0f*ay; a1.x = -2.0f*bx; a1.y = -2.0f*by; }
        else           { a0.x = -2.0f*az; a0.y = 1.0f;     a1.x = -2.0f*bz; a1.y = 1.0f;     }
    }

    // Register-resident sorted top-k (ascending key). All indices compile-time.
    float bk[KNN_K];
    int   bi[KNN_K];
    #pragma unroll
    for (int t = 0; t < KNN_K; ++t) { bk[t] = __builtin_inff(); bi[t] = 0; }

    const int ntiles = (n + 15) / 16;
    for (int jt = 0; jt < ntiles; ++jt) {
        const int j0 = jt * 16;

        // B tile (4x16): column j = j0+col holds (x_j, y_j, z_j, ||x_j||^2).
        const int j  = j0 + col;
        const int jj = j < n ? j : n - 1;
        v2f b;
        if (half == 0) {
            b.x = pos[3*jj+0];
            b.y = pos[3*jj+1];
        } else {
            const float jx = pos[3*jj+0], jy = pos[3*jj+1], jz = pos[3*jj+2];
            b.x = jz;
            b.y = jx*jx + jy*jy + jz*jz;
        }

        // D(i,j) = ||x_j||^2 - 2 * <x_i, x_j>  == the ranking key directly.
        v8f zacc = {0.f,0.f,0.f,0.f,0.f,0.f,0.f,0.f};
        v8f d0 = __builtin_amdgcn_wmma_f32_16x16x4_f32(
            false, a0, false, b, (short)0, zacc, false, false);
        v8f d1 = __builtin_amdgcn_wmma_f32_16x16x4_f32(
            false, a1, false, b, (short)0, zacc, false, false);

        // Transpose via LDS: lane(half,col) holds rows half*8..half*8+7 of column col.
        #pragma unroll
        for (int m = 0; m < 8; ++m) {
            tile[0][half*8 + m][col] = d0[m];
            tile[1][half*8 + m][col] = d1[m];
        }
        __syncthreads();

        // My center's row of 16 keys is exactly tile[half][col][0..15] (contiguous).
        const float4* rp = (const float4*)(&tile[half][col][0]);
        #pragma unroll
        for (int q = 0; q < 4; ++q) {
            const float4 v = rp[q];
            const float kv[4] = {v.x, v.y, v.z, v.w};
            #pragma unroll
            for (int e = 0; e < 4; ++e) {
                const int   jc  = j0 + q*4 + e;
                const float key = kv[e];
                // exclude self-loop by index (its fused key is -||x||^2, not ~0)
                if (jc < n && jc != c && key < bk[KNN_K-1]) {
                    // unrolled sorted insertion, evaluated high->low so bk[p-1]
                    // is read before being overwritten
                    #pragma unroll
                    for (int p = KNN_K-1; p >= 0; --p) {
                        const bool here   = key < bk[p];
                        const bool before = (p > 0) && (key < bk[p-1]);
                        const float nd = before ? bk[p-1] : key;
                        const int   ni = before ? bi[p-1] : jc;
                        if (here) { bk[p] = nd; bi[p] = ni; }
                    }
                }
            }
        }
        __syncthreads();   // protect LDS tile before next iteration's stores
    }

    if (c < n) {
        const long base = (long)c * KNN_K;
        #pragma unroll
        for (int t = 0; t < KNN_K; ++t) {
            ei_src[base + t] = bi[t];                              // source = neighbor idx
            ei_dst[base + t] = c;                                  // dest   = center idx
            ed[base + t]     = sqrtf(fmaxf(x2i + bk[t], 0.0f));    // Euclidean distance
        }
    }
}

extern "C" void kernel_launch(void* const* d_in, const int* in_sizes, int n_in,
                              void* d_out, int out_size, void* d_ws, size_t ws_size,
                              hipStream_t stream) {
    (void)n_in; (void)out_size; (void)d_ws; (void)ws_size;
    const float* pos = (const float*)d_in[0];
    const int n = in_sizes[0] / 3;           // pos is [N,3] float32
    const int k = KNN_K;                      // reference k = 17

    // d_out layout: edge_index int32 [2, N*k] then edge_dist float [N*k]
    int*   ei_src = (int*)d_out;
    int*   ei_dst = ei_src + (size_t)n * k;
    float* ed     = (float*)(ei_src + (size_t)2 * n * k);

    const int blocks = (n + 31) / 32;         // one wave32 per 32 centers
    knn_wmma_kernel<<<blocks, 32, 0, stream>>>(pos, n, ei_src, ei_dst, ed);
}